// WaveletLayer_31035433681584
// MI455X (gfx1250) — compile-verified
//
#include <hip/hip_runtime.h>

// ---------------------------------------------------------------------------
// WaveletLayer (db2 DWT -> scale cD by weight[0] -> db2 IDWT -> leaky_relu)
// as a banded linear operator z = A x per row, tiled onto
// V_WMMA_F32_16X16X4_F32 (gfx1250 / CDNA5, wave32).
//
// Kernel 1: build the 32 distinct (20 x 16) column tiles of A into d_ws (40KB).
// Kernel 2: per workgroup: 16 rows staged into LDS (stride 524 floats ->
//           conflict-free transposed A reads), 8 waves x 4 tiles x 5 WMMAs,
//           fused leaky-ReLU on the D matrix, coalesced stores.
// ---------------------------------------------------------------------------

typedef __attribute__((ext_vector_type(2))) float v2f;
typedef __attribute__((ext_vector_type(8))) float v8f;

#define D_LEN      512
#define NHALF      257          // D/2 + 1
#define LDS_STRIDE 524          // 524 % 64 == 12 -> 16 rows hit 16 distinct banks
#define NTILES     32           // 512 / 16 output-column tiles
#define KWIN       20           // input window per 16-output tile (pad of band)
#define WSZ        (NTILES * KWIN * 16)   // 10240 floats = 40KB in d_ws

// db2 filter banks (f32)
#define DL0 (-0.12940952255092145f)
#define DL1 ( 0.22414386804185735f)
#define DL2 ( 0.836516303737469f)
#define DL3 ( 0.48296291314469025f)
#define DH0 (-0.48296291314469025f)
#define DH1 ( 0.836516303737469f)
#define DH2 (-0.22414386804185735f)
#define DH3 (-0.12940952255092145f)
// REC_LO / REC_HI
#define RL0 ( 0.48296291314469025f)
#define RL1 ( 0.836516303737469f)
#define RL2 ( 0.22414386804185735f)
#define RL3 (-0.12940952255092145f)
#define RH0 (-0.12940952255092145f)
#define RH1 (-0.22414386804185735f)
#define RH2 ( 0.836516303737469f)
#define RH3 (-0.48296291314469025f)

// DLr[t] = DEC_LO[3-t], DHr[t] = DEC_HI[3-t]  (coefficient hitting x[2m-2+t])
__device__ __constant__ float c_DLr[4] = { DL3, DL2, DL1, DL0 };
__device__ __constant__ float c_DHr[4] = { DH3, DH2, DH1, DH0 };

// ---------------------------------------------------------------------------
// Kernel 1: W[c][kk][nn] = A[c*16+nn][c*16-2+kk]
//
// With cA[k] = sum_t x[2k-2+t]*DEC_LO[3-t] (interior; symmetric-ext edges at
// k==0 and k==256), d[k] = w[k]*cD[k], and
//   z[2m]   = RL2*cA[m] + RL0*cA[m+1] + RH2*d[m] + RH0*d[m+1]
//   z[2m+1] = RL3*cA[m] + RL1*cA[m+1] + RH3*d[m] + RH1*d[m+1]
// ---------------------------------------------------------------------------
__global__ __launch_bounds__(256)
void build_w_kernel(const float* __restrict__ w, float* __restrict__ Wg) {
    int e = blockIdx.x * 256 + threadIdx.x;
    if (e >= WSZ) return;
    int nn = e & 15;
    int kk = (e >> 4) % KWIN;
    int c  = e / (KWIN * 16);
    int n  = c * 16 + nn;            // output column
    int i  = c * 16 - 2 + kk;        // input column (window base = c*16-2)

    float acc = 0.0f;
    if (i >= 0 && i < D_LEN) {
        int m   = n >> 1;
        int odd = n & 1;
        float RLa = odd ? RL3 : RL2, RHa = odd ? RH3 : RH2;  // weight on idx m
        float RLb = odd ? RL1 : RL0, RHb = odd ? RH1 : RH0;  // weight on idx m+1

        // contribution through cA[m] / d[m]
        if (m == 0) {
            // cA[0] = (DL2+DL1)*x0 + (DL3+DL0)*x1 ; same pattern for cD with DH
            if (i == 0)      acc += RLa * (DL2 + DL1) + RHa * w[0] * (DH2 + DH1);
            else if (i == 1) acc += RLa * (DL3 + DL0) + RHa * w[0] * (DH3 + DH0);
        } else {
            int t = i - (2 * m - 2);
            if (t >= 0 && t < 4)
                acc += RLa * c_DLr[t] + RHa * w[m] * c_DHr[t];
        }
        // contribution through cA[m+1] / d[m+1]
        int mp = m + 1;
        if (mp == NHALF - 1) {       // k == 256 edge: uses x[510], x[511]
            if (i == 510)      acc += RLb * (DL3 + DL0) + RHb * w[256] * (DH3 + DH0);
            else if (i == 511) acc += RLb * (DL2 + DL1) + RHb * w[256] * (DH2 + DH1);
        } else {
            int t = i - 2 * m;
            if (t >= 0 && t < 4)
                acc += RLb * c_DLr[t] + RHb * w[mp] * c_DHr[t];
        }
    }
    Wg[e] = acc;
}

// ---------------------------------------------------------------------------
// Kernel 2: 16 rows per workgroup, 8 waves, each wave does 4 column tiles.
// ---------------------------------------------------------------------------
__global__ __launch_bounds__(256)
void wavelet_wmma_kernel(const float* __restrict__ x,
                         const float* __restrict__ Wg,
                         float* __restrict__ out) {
    __shared__ float xs[16 * LDS_STRIDE];    // 33,536 B

    const int  tid  = threadIdx.x;
    const int  lane = tid & 31;              // wave32
    const int  wave = tid >> 5;              // 0..7
    const long r0   = (long)blockIdx.x * 16; // first batch row of this block

    // zero the halo (left cols i=-4..-1 -> idx 0..3; right i=512.. -> idx 516..523)
    if (tid < 16) {
        int b = tid * LDS_STRIDE;
        xs[b + 0] = 0.f; xs[b + 1] = 0.f; xs[b + 2] = 0.f; xs[b + 3] = 0.f;
#pragma unroll
        for (int h = 516; h < LDS_STRIDE; ++h) xs[b + h] = 0.f;
    }

    // stage 16 x 512 f32 into LDS, padded stride, 16B-aligned b128 traffic
    const float4* xv = (const float4*)(x + r0 * D_LEN);
#pragma unroll
    for (int s = 0; s < 8; ++s) {
        int idx = tid + 256 * s;             // 0..2047 float4 tiles
        int row = idx >> 7;                  // 0..15
        int c4  = idx & 127;                 // float4 within row
        float4 v = xv[row * 128 + c4];
        *(float4*)&xs[row * LDS_STRIDE + 4 + c4 * 4] = v;
    }
    __syncthreads();

    // WMMA operand lane roles (V_WMMA_F32_16X16X4_F32):
    //  A (16x4):  VGPR0: lanes0-15 -> (M=lane,K=0), lanes16-31 -> (M=lane-16,K=2)
    //             VGPR1: K=1 / K=3
    //  B (4x16):  VGPR0: lanes0-15 -> (K=0,N=lane), lanes16-31 -> (K=2,N=lane-16)
    //             VGPR1: K=1 / K=3
    //  C/D (16x16): VGPR v: lanes0-15 -> (M=v,N), lanes16-31 -> (M=v+8,N)
    const int mrow  = lane & 15;
    const int khalf = (lane >> 4) << 1;      // 0 or 2
    const int ncol  = lane & 15;
    const int mh    = (lane >> 4) << 3;      // 0 or 8

#pragma unroll
    for (int j = 0; j < 4; ++j) {
        const int c = wave * 4 + j;          // column tile 0..31

        v8f acc = {0.f, 0.f, 0.f, 0.f, 0.f, 0.f, 0.f, 0.f};

        // LDS float index of A(M=mrow, K=khalf) for q=0; window base c*16-2, halo +4
        const int abase = mrow * LDS_STRIDE + c * 16 + 2 + khalf;
        // B tile base for this lane
        const float* wb = Wg + c * (KWIN * 16) + khalf * 16 + ncol;

#if __has_builtin(__builtin_amdgcn_wmma_f32_16x16x4_f32)
#pragma unroll
        for (int q = 0; q < 5; ++q) {        // K = 20 = 5 x 4
            v2f a, b;
            a.x = xs[abase + 4 * q];
            a.y = xs[abase + 4 * q + 1];
            b.x = wb[(4 * q) * 16];
            b.y = wb[(4 * q + 1) * 16];
            acc = __builtin_amdgcn_wmma_f32_16x16x4_f32(
                false, a, false, b, (short)0, acc, false, false);
        }
#else
        // VALU fallback with identical D layout (keeps kernel compiling if the
        // f32x4 WMMA builtin name/signature differs on this toolchain).
#pragma unroll
        for (int v = 0; v < 8; ++v) {
            float s = 0.f;
            const int M = mh + v;
#pragma unroll
            for (int k = 0; k < KWIN; ++k)
                s += xs[M * LDS_STRIDE + c * 16 + 2 + k] *
                     Wg[c * (KWIN * 16) + k * 16 + ncol];
            acc[v] = s;
        }
#endif

        // D -> leaky_relu -> out ; VGPR v holds rows (mh + v)
        float* op = out + (r0 + mh) * D_LEN + c * 16 + ncol;
#pragma unroll
        for (int v = 0; v < 8; ++v) {
            float z = acc[v];
            z = (z >= 0.f) ? z : 0.01f * z;
            op[(long)v * D_LEN] = z;
        }
    }
}

// ---------------------------------------------------------------------------
extern "C" void kernel_launch(void* const* d_in, const int* in_sizes, int n_in,
                              void* d_out, int out_size, void* d_ws, size_t ws_size,
                              hipStream_t stream) {
    const float* x      = (const float*)d_in[0];   // (B, 512) f32
    const float* weight = (const float*)d_in[1];   // (2, 257) f32; row 0 used
    float*       out    = (float*)d_out;           // (B, 512) f32
    float*       Wg     = (float*)d_ws;            // 40KB operator tiles

    const int B = in_sizes[0] / D_LEN;             // 65536

    build_w_kernel<<<(WSZ + 255) / 256, 256, 0, stream>>>(weight, Wg);
    wavelet_wmma_kernel<<<B / 16, 256, 0, stream>>>(x, Wg, out);
}